// MultiDense_53781580480938
// MI455X (gfx1250) — compile-verified
//
#include <hip/hip_runtime.h>

typedef float v2f __attribute__((ext_vector_type(2)));
typedef float v8f __attribute__((ext_vector_type(8)));

#define J_DIM   32
#define IN_F    128
#define OUT_F   128
#define PSTRIDE (IN_F + 1)   // 129 floats per (node, out_feature) row: 128 weights + bias

// One block per sample i. 256 threads = 8 waves.
// Wave w computes output features [16w, 16w+16) for all 32 rows (two 16x16 M-tiles).
__global__ __launch_bounds__(256) void multidense_wmma_f32(
    const float* __restrict__ x,       // (I, 32, 128)
    const int*   __restrict__ inds,    // (I,)
    const float* __restrict__ params,  // (4096, 128, 129)
    float*       __restrict__ out)     // (I, 32, 128)
{
    __shared__ float xs[J_DIM * IN_F];                    // 16 KB x-tile

    const int i    = blockIdx.x;
    const int tid  = threadIdx.x;
    const int lane = tid & 31;
    const int wave = tid >> 5;                            // 0..7 -> N-tile
    const int l16  = lane & 15;
    const int half = lane >> 4;                           // 0 or 1 (K sub-offset)

    // ---- stage x[i] (32x128 f32 = 4096 floats) into LDS, coalesced float4 ----
    {
        const float4* xg  = (const float4*)(x + (size_t)i * (J_DIM * IN_F));
        float4*       xs4 = (float4*)xs;
#pragma unroll
        for (int t = 0; t < 4; ++t)
            xs4[tid + 256 * t] = xg[tid + 256 * t];
    }

    const int    node = inds[i];
    const float* W    = params + (size_t)node * OUT_F * PSTRIDE;

    // Per-lane base pointers.
    const int    n0    = wave * 16;
    const float* wrow  = W  + (size_t)(n0 + l16) * PSTRIDE;   // this lane's W row (its N column)

    // Bias for this lane's output feature (same for every accumulator element) --
    // fold it into the WMMA C initialization instead of a VALU epilogue.
    const float bias = wrow[IN_F];
    v8f acc0 = {bias, bias, bias, bias, bias, bias, bias, bias};
    v8f acc1 = acc0;

    __syncthreads();

    const float* xrow0 = xs + l16 * IN_F;                     // M-tile 0, row = l16
    const float* xrow1 = xs + (16 + l16) * IN_F;              // M-tile 1

    // K loop: 128 / 4 = 32 steps, 2 WMMAs each (both M-tiles share B).
#pragma unroll
    for (int k = 0; k < IN_F; k += 4) {
        const int ko = k + 2 * half;                          // lanes 16-31 take K+2,K+3

        // A fragments from LDS: 8B-aligned (ko even, row stride 128) -> ds_load_2addr_b64
        const v2f a0 = *(const v2f*)(xrow0 + ko);
        const v2f a1 = *(const v2f*)(xrow1 + ko);

        // B fragment from global W row (stride 129 floats); compiler emits global_load_b64
        v2f b;
        b.x = wrow[ko];
        b.y = wrow[ko + 1];

        acc0 = __builtin_amdgcn_wmma_f32_16x16x4_f32(
            /*neg_a=*/false, a0, /*neg_b=*/false, b,
            /*c_mod=*/(short)0, acc0, /*reuse_a=*/false, /*reuse_b=*/false);
        acc1 = __builtin_amdgcn_wmma_f32_16x16x4_f32(
            false, a1, false, b, (short)0, acc1, false, false);
    }

    // D layout: VGPR r -> row (8*half + r); col = n0 + l16. Bias already in acc.
    float*    op      = out + (size_t)i * (J_DIM * OUT_F) + n0 + l16;
    const int rowbase = half * 8;
#pragma unroll
    for (int r = 0; r < 8; ++r) {
        op[(size_t)(rowbase + r) * OUT_F]      = acc0[r];
        op[(size_t)(16 + rowbase + r) * OUT_F] = acc1[r];
    }
}

extern "C" void kernel_launch(void* const* d_in, const int* in_sizes, int n_in,
                              void* d_out, int out_size, void* d_ws, size_t ws_size,
                              hipStream_t stream)
{
    const float* x      = (const float*)d_in[0];
    const int*   inds   = (const int*)d_in[1];
    const float* params = (const float*)d_in[2];
    float*       out    = (float*)d_out;

    const int I = in_sizes[0] / (J_DIM * IN_F);   // 8192

    multidense_wmma_f32<<<dim3(I), dim3(256), 0, stream>>>(x, inds, params, out);
}